// ForwardDeformer_50818053047085
// MI455X (gfx1250) — compile-verified
//
#include <hip/hip_runtime.h>
#include <hip/hip_bf16.h>

// ---------------------------------------------------------------------------
// ForwardDeformer for MI455X (gfx1250, wave32).
//   K1: bone-weight GEMM via V_WMMA_F32_16X16X4_F32 + per-voxel 3x3 inverse
//   K2: invert the 9 init-bone transforms
//   K3: 10 Newton iterations with trilinear sampling of interleaved voxel
//       records (3x float4 per corner), LDS-based duplicate suppression
// ---------------------------------------------------------------------------

#define RES      64
#define M_VOX    (RES * RES * RES)   // 262144 voxels
#define NBONES   24
#define NPTS     131072
#define NINIT    9
#define NITER    10
#define CVG      1.0e-4f
#define DVG      0.5f
#define DUP_EPS2 1.0e-6f

typedef __attribute__((ext_vector_type(2))) float v2f;
typedef __attribute__((ext_vector_type(8))) float v8f;

// ---------------------------------------------------------------------------
// Kernel 1: T[m] = sum_j wts[j][m] * tfs[j]  (M x 24 @ 24 x 16 f32 GEMM via
// WMMA f32 16x16x4), then vox_d = R p + t and Rinv = inv(R).
// Voxel record (12 floats = 3 float4): [dx dy dz i00][i01 i02 i10 i11][i12 i20 i21 i22]
// ---------------------------------------------------------------------------
__global__ __launch_bounds__(256) void build_voxels_kernel(
    const float* __restrict__ tfs,     // [24][16] row-major 4x4
    const float* __restrict__ lbs,     // [24][M]
    const float* __restrict__ offset,  // [3]
    const float* __restrict__ scale,   // [3]
    float* __restrict__ vox)           // [M][12]
{
    __shared__ float lds[8][32][16];   // per-wave 32 voxels x 16 transform elems

    const int tid  = threadIdx.x;
    const int wave = tid >> 5;
    const int lane = tid & 31;
    const int hl   = lane >> 4;        // half-wave select (0 or 1)
    const int l16  = lane & 15;
    const int vbase = blockIdx.x * 256 + wave * 32;

    // Two 16x16 f32 accumulator tiles: voxels [vbase..+15], [vbase+16..+31].
    v8f c0 = {};
    v8f c1 = {};

    // K = 24 bones, 6 steps of K=4.
    // A 16x4 f32 layout: lanes 0-15 rows M=0..15, VGPR0 = K {0|2}, VGPR1 = K {1|3}.
    // B 4x16 f32 layout (mirror): lanes hold N=0..15, VGPR0 = K {0|2}, VGPR1 = K {1|3}.
    for (int ks = 0; ks < 6; ++ks) {
        const int ka = ks * 4 + hl * 2;
        v2f a0, a1, b;
        a0.x = lbs[(ka + 0) * M_VOX + vbase + l16];
        a0.y = lbs[(ka + 1) * M_VOX + vbase + l16];
        a1.x = lbs[(ka + 0) * M_VOX + vbase + 16 + l16];
        a1.y = lbs[(ka + 1) * M_VOX + vbase + 16 + l16];
        b.x  = tfs[(ka + 0) * 16 + l16];
        b.y  = tfs[(ka + 1) * 16 + l16];
        c0 = __builtin_amdgcn_wmma_f32_16x16x4_f32(false, a0, false, b,
                                                   (short)0, c0, false, false);
        c1 = __builtin_amdgcn_wmma_f32_16x16x4_f32(false, a1, false, b,
                                                   (short)0, c1, false, false);
    }

    // C/D layout: VGPR r, lanes 0-15 -> row M=r, lanes 16-31 -> row M=8+r; col N = l16.
    #pragma unroll
    for (int r = 0; r < 8; ++r) {
        lds[wave][hl * 8 + r][l16]      = c0[r];
        lds[wave][16 + hl * 8 + r][l16] = c1[r];
    }
    __syncthreads();

    // Each lane now owns one voxel's full 16-element transform.
    float t[16];
    #pragma unroll
    for (int k = 0; k < 16; ++k) t[k] = lds[wave][lane][k];

    const int v  = vbase + lane;
    const int xg = v & 63;
    const int yg = (v >> 6) & 63;
    const int zg = v >> 12;

    const float px = (-1.0f + 2.0f * (float)xg / 63.0f) / scale[0] - offset[0];
    const float py = (-1.0f + 2.0f * (float)yg / 63.0f) / scale[1] - offset[1];
    const float pz = (-1.0f + 2.0f * (float)zg / 63.0f) / scale[2] - offset[2];

    const float r00 = t[0], r01 = t[1], r02 = t[2],  tx = t[3];
    const float r10 = t[4], r11 = t[5], r12 = t[6],  ty = t[7];
    const float r20 = t[8], r21 = t[9], r22 = t[10], tz = t[11];

    const float dx = r00 * px + r01 * py + r02 * pz + tx;
    const float dy = r10 * px + r11 * py + r12 * pz + ty;
    const float dz = r20 * px + r21 * py + r22 * pz + tz;

    const float det = r00 * (r11 * r22 - r12 * r21)
                    - r01 * (r10 * r22 - r12 * r20)
                    + r02 * (r10 * r21 - r11 * r20);
    const float rd = 1.0f / det;
    const float i00 = (r11 * r22 - r12 * r21) * rd;
    const float i01 = (r02 * r21 - r01 * r22) * rd;
    const float i02 = (r01 * r12 - r02 * r11) * rd;
    const float i10 = (r12 * r20 - r10 * r22) * rd;
    const float i11 = (r00 * r22 - r02 * r20) * rd;
    const float i12 = (r02 * r10 - r00 * r12) * rd;
    const float i20 = (r10 * r21 - r11 * r20) * rd;
    const float i21 = (r01 * r20 - r00 * r21) * rd;
    const float i22 = (r00 * r11 - r01 * r10) * rd;

    float4* vout = reinterpret_cast<float4*>(vox) + (size_t)v * 3;
    vout[0] = make_float4(dx,  dy,  dz,  i00);
    vout[1] = make_float4(i01, i02, i10, i11);
    vout[2] = make_float4(i12, i20, i21, i22);
}

// ---------------------------------------------------------------------------
// Kernel 2: tf_inv[i] = inv(tfs[init_bones[i]]) stored as [Rinv | -Rinv t].
// ---------------------------------------------------------------------------
__global__ void invert_init_kernel(const float* __restrict__ tfs,
                                   const int* __restrict__ init_bones,
                                   float* __restrict__ tf_inv)
{
    const int i = threadIdx.x;
    if (i >= NINIT) return;
    const float* T = tfs + init_bones[i] * 16;
    const float r00 = T[0], r01 = T[1], r02 = T[2],  tx = T[3];
    const float r10 = T[4], r11 = T[5], r12 = T[6],  ty = T[7];
    const float r20 = T[8], r21 = T[9], r22 = T[10], tz = T[11];
    const float det = r00 * (r11 * r22 - r12 * r21)
                    - r01 * (r10 * r22 - r12 * r20)
                    + r02 * (r10 * r21 - r11 * r20);
    const float rd = 1.0f / det;
    const float i00 = (r11 * r22 - r12 * r21) * rd;
    const float i01 = (r02 * r21 - r01 * r22) * rd;
    const float i02 = (r01 * r12 - r02 * r11) * rd;
    const float i10 = (r12 * r20 - r10 * r22) * rd;
    const float i11 = (r00 * r22 - r02 * r20) * rd;
    const float i12 = (r02 * r10 - r00 * r12) * rd;
    const float i20 = (r10 * r21 - r11 * r20) * rd;
    const float i21 = (r01 * r20 - r00 * r21) * rd;
    const float i22 = (r00 * r11 - r01 * r10) * rd;
    float* o = tf_inv + i * 12;
    o[0] = i00; o[1]  = i01; o[2]  = i02; o[3]  = -(i00 * tx + i01 * ty + i02 * tz);
    o[4] = i10; o[5]  = i11; o[6]  = i12; o[7]  = -(i10 * tx + i11 * ty + i12 * tz);
    o[8] = i20; o[9]  = i21; o[10] = i22; o[11] = -(i20 * tx + i21 * ty + i22 * tz);
}

// ---------------------------------------------------------------------------
// Kernel 3: Newton iterations + duplicate suppression.
// Block = 288 threads = 9 waves; wave index = init i, lane = point within tile.
// ---------------------------------------------------------------------------
__global__ __launch_bounds__(288) void newton_iterate_kernel(
    const float* __restrict__ xd,      // [N][3]
    const float* __restrict__ vox,     // [M][12]
    const float* __restrict__ tf_inv,  // [9][12]
    const float* __restrict__ offset,  // [3]
    const float* __restrict__ scale,   // [3]
    float* __restrict__ out_xc,        // [N*9*3]
    float* __restrict__ out_mask,      // [N*9]
    float* __restrict__ out_J)         // [N*9*9]
{
    __shared__ float sxc[NINIT][32][3];
    __shared__ float svalid[NINIT][32];

    const int tid  = threadIdx.x;
    const int i    = tid >> 5;       // init index 0..8
    const int lane = tid & 31;
    const int n    = blockIdx.x * 32 + lane;

    const float xdx = xd[n * 3 + 0];
    const float xdy = xd[n * 3 + 1];
    const float xdz = xd[n * 3 + 2];

    const float* Ti = tf_inv + i * 12;
    float xcx = Ti[0] * xdx + Ti[1] * xdy + Ti[2] * xdz + Ti[3];
    float xcy = Ti[4] * xdx + Ti[5] * xdy + Ti[6] * xdz + Ti[7];
    float xcz = Ti[8] * xdx + Ti[9] * xdy + Ti[10] * xdz + Ti[11];

    float Jl[9] = {1.f, 0.f, 0.f, 0.f, 1.f, 0.f, 0.f, 0.f, 1.f};
    bool conv = false, dive = false;

    const float offx = offset[0], offy = offset[1], offz = offset[2];
    const float scx = scale[0], scy = scale[1], scz = scale[2];
    const float4* v4 = reinterpret_cast<const float4*>(vox);

    for (int it = 0; it < NITER; ++it) {
        const float nx = (xcx + offx) * scx;
        const float ny = (xcy + offy) * scy;
        const float nz = (xcz + offz) * scz;
        const float ix = (nx + 1.0f) * 0.5f * 63.0f;
        const float iy = (ny + 1.0f) * 0.5f * 63.0f;
        const float iz = (nz + 1.0f) * 0.5f * 63.0f;
        const float fx = floorf(ix), fy = floorf(iy), fz = floorf(iz);
        const float wx = ix - fx, wy = iy - fy, wz = iz - fz;

        float dd0 = 0.f, dd1 = 0.f, dd2 = 0.f;
        float JJ[9] = {0.f, 0.f, 0.f, 0.f, 0.f, 0.f, 0.f, 0.f, 0.f};

        #pragma unroll
        for (int cz = 0; cz < 2; ++cz) {
            #pragma unroll
            for (int cy = 0; cy < 2; ++cy) {
                #pragma unroll
                for (int cx = 0; cx < 2; ++cx) {
                    const float xf = fx + (float)cx;
                    const float yf = fy + (float)cy;
                    const float zf = fz + (float)cz;
                    const bool inb = (xf >= 0.f) & (xf <= 63.f) &
                                     (yf >= 0.f) & (yf <= 63.f) &
                                     (zf >= 0.f) & (zf <= 63.f);
                    float w = (cx ? wx : 1.0f - wx) *
                              (cy ? wy : 1.0f - wy) *
                              (cz ? wz : 1.0f - wz);
                    w = inb ? w : 0.0f;
                    // NaN-safe clamp (fmaxf/fminf return the non-NaN operand)
                    const int xi = (int)fminf(fmaxf(xf, 0.f), 63.f);
                    const int yi = (int)fminf(fmaxf(yf, 0.f), 63.f);
                    const int zi = (int)fminf(fmaxf(zf, 0.f), 63.f);
                    const int m = ((zi * 64 + yi) * 64 + xi) * 3;
                    const float4 f0 = v4[m + 0];
                    const float4 f1 = v4[m + 1];
                    const float4 f2 = v4[m + 2];
                    dd0  += w * f0.x; dd1  += w * f0.y; dd2  += w * f0.z;
                    JJ[0] += w * f0.w; JJ[1] += w * f1.x; JJ[2] += w * f1.y;
                    JJ[3] += w * f1.z; JJ[4] += w * f1.w; JJ[5] += w * f2.x;
                    JJ[6] += w * f2.y; JJ[7] += w * f2.z; JJ[8] += w * f2.w;
                }
            }
        }

        const float gx = dd0 - xdx, gy = dd1 - xdy, gz = dd2 - xdz;
        const float err = sqrtf(gx * gx + gy * gy + gz * gz);
        const bool active = !(conv || dive);
        const bool c_new  = active && (err < CVG);
        const bool d_new  = active && (err > DVG);
        const bool step   = active && (err >= CVG) && (err <= DVG);
        if (step) {
            xcx -= JJ[0] * gx + JJ[1] * gy + JJ[2] * gz;
            xcy -= JJ[3] * gx + JJ[4] * gy + JJ[5] * gz;
            xcz -= JJ[6] * gx + JJ[7] * gy + JJ[8] * gz;
        }
        if (active) {
            #pragma unroll
            for (int k = 0; k < 9; ++k) Jl[k] = JJ[k];
        }
        conv = conv || c_new;
        dive = dive || d_new;
    }

    // Cross-init duplicate suppression via LDS.
    sxc[i][lane][0] = xcx;
    sxc[i][lane][1] = xcy;
    sxc[i][lane][2] = xcz;
    svalid[i][lane] = conv ? 1.0f : 0.0f;
    __syncthreads();

    bool dup = false;
    for (int j = 0; j < i; ++j) {
        const float ax = xcx - sxc[j][lane][0];
        const float ay = xcy - sxc[j][lane][1];
        const float az = xcz - sxc[j][lane][2];
        dup = dup || ((svalid[j][lane] != 0.0f) &&
                      (ax * ax + ay * ay + az * az < DUP_EPS2));
    }

    const int pi = n * NINIT + i;
    out_xc[pi * 3 + 0] = xcx;
    out_xc[pi * 3 + 1] = xcy;
    out_xc[pi * 3 + 2] = xcz;
    out_mask[pi] = (conv && !dup) ? 1.0f : 0.0f;
    #pragma unroll
    for (int k = 0; k < 9; ++k) out_J[pi * 9 + k] = Jl[k];
}

// ---------------------------------------------------------------------------
extern "C" void kernel_launch(void* const* d_in, const int* in_sizes, int n_in,
                              void* d_out, int out_size, void* d_ws, size_t ws_size,
                              hipStream_t stream)
{
    const float* xd        = (const float*)d_in[0];  // [1,131072,3]
    const float* tfs       = (const float*)d_in[1];  // [1,24,4,4]
    const float* lbs       = (const float*)d_in[2];  // [1,24,64,64,64]
    const float* offset    = (const float*)d_in[3];  // [1,1,3]
    const float* scale     = (const float*)d_in[4];  // [1,1,3]
    const int*   init_bone = (const int*)d_in[5];    // [9]

    float* ws     = (float*)d_ws;
    float* vox    = ws;                              // [M][12] floats (12.6 MB)
    float* tf_inv = ws + (size_t)M_VOX * 12;         // [9][12] floats

    float* out      = (float*)d_out;
    float* out_xc   = out;                                  // N*9*3
    float* out_mask = out + (size_t)NPTS * NINIT * 3;       // N*9
    float* out_J    = out + (size_t)NPTS * NINIT * 4;       // N*9*9

    build_voxels_kernel<<<M_VOX / 256, 256, 0, stream>>>(tfs, lbs, offset, scale, vox);
    invert_init_kernel<<<1, 16, 0, stream>>>(tfs, init_bone, tf_inv);
    newton_iterate_kernel<<<NPTS / 32, 288, 0, stream>>>(xd, vox, tf_inv, offset, scale,
                                                         out_xc, out_mask, out_J);
}